// Correlation_6571299963137
// MI455X (gfx1250) — compile-verified
//
#include <hip/hip_runtime.h>

typedef float v2f __attribute__((ext_vector_type(2)));
typedef float v8f __attribute__((ext_vector_type(8)));

constexpr int kB = 8;
constexpr int kC = 256;
constexpr int kH = 64;
constexpr int kW = 128;
constexpr int kD = 21;            // 21x21 displacement grid, d = dyIdx*21 + dxIdx
constexpr int kKC = 16;           // channels per LDS stage (double buffered)
constexpr int kNStage = kC / kKC; // 16
constexpr int kSA = 136;          // sA row stride (128 cols + 8 pad -> bank-split halves)
constexpr int kSB = 184;          // sB row stride (176 cols + 8 pad)
constexpr int kXpOff = 24;        // sB col s maps to x' = s - 24, s in [0,176)
constexpr int kAbuf = kKC * kSA;  // floats per sA buffer
constexpr int kBbuf = kKC * kSB;  // floats per sB buffer

#define USE_ASYNC_ASM 1

// Copy 16B global -> LDS per lane. Async path: CDNA5 GLOBAL_LOAD_ASYNC_TO_LDS_B128
// (ASYNCcnt-tracked DMA). The VDST operand is a VGPR holding the LDS byte address;
// low 32 bits of a generic pointer to a __shared__ object are the AS3 offset.
__device__ __forceinline__ void cp16(const float* g, float* l) {
#if USE_ASYNC_ASM
  asm volatile("global_load_async_to_lds_b128 %0, %1, off"
               :
               : "v"((unsigned)(size_t)l), "v"(g)
               : "memory");
#else
  *(float4*)l = *(const float4*)g;
#endif
}

__device__ __forceinline__ void wait_async() {
#if USE_ASYNC_ASM
  asm volatile("s_wait_asynccnt 0" ::: "memory");
#endif
}

// One block = (b, y, dyIdx). 8 waves, wave w owns pixel tile x0 = 16*w.
// Band GEMM: P[x, x'] = sum_c in1[c,x] * in2[c,x'], x' tiles at x0-24+16j, j=0..3.
__global__ __launch_bounds__(256) void corr_wmma_kernel(
    const float* __restrict__ in1, const float* __restrict__ in2,
    float* __restrict__ out)
{
  __shared__ float sA[2 * kAbuf];
  __shared__ float sB[2 * kBbuf];
  __shared__ float sOut[kD * kW];

  const int y     = blockIdx.x;
  const int dyIdx = blockIdx.y;
  const int b     = blockIdx.z;
  const int y2    = y + (2 * dyIdx - 20);
  const int tid   = threadIdx.x;
  const bool rowValid = (y2 >= 0) && (y2 < kH);   // block-uniform

  if (rowValid) {
    const int lane = tid & 31;          // wave32
    const int x0   = (tid >> 5) << 4;   // wave's M tile base
    const int m    = lane & 15;
    const int hi   = lane >> 4;
    const int kk   = hi << 1;           // A/B frag: vgpr0 holds K=kk, vgpr1 K=kk+1

    const float* g1 = in1 + ((size_t)(b * kC) * kH + y)  * kW;  // + c*kH*kW
    const float* g2 = in2 + ((size_t)(b * kC) * kH + y2) * kW;

    // ---- zero sB borders once (x' outside [0,128)); disjoint from async region ----
    const float4 z4 = make_float4(0.f, 0.f, 0.f, 0.f);
    for (int f = tid; f < 2 * kKC * 12; f += 256) {   // 12 border float4 per row
      int buf = f / (kKC * 12);
      int rem = f - buf * (kKC * 12);
      int row = rem / 12;
      int idx = rem - row * 12;
      int s4  = (idx < 6) ? (idx << 2) : (152 + ((idx - 6) << 2));
      *(float4*)(sB + buf * kBbuf + row * kSB + s4) = z4;
    }

    // ---- async-stage one 16-channel K chunk into buffer `buf` ----
    auto issue_stage = [&](int s, int buf) {
      const int cb = s * kKC;
      #pragma unroll
      for (int k = 0; k < 2; ++k) {               // 512 float4 per array, 2/thread
        int f   = tid + k * 256;
        int row = f >> 5;
        int col = (f & 31) << 2;
        const size_t go = (size_t)(cb + row) * (kH * kW) + col;
        cp16(g1 + go, sA + buf * kAbuf + row * kSA + col);
        cp16(g2 + go, sB + buf * kBbuf + row * kSB + col + kXpOff);
      }
    };

    v8f acc0 = {}, acc1 = {}, acc2 = {}, acc3 = {};
    const int aBase = kk * kSA + x0 + m;
    const int bBase = kk * kSB + x0 + m;          // + 16*j per N tile (s = x0+16j+m)

    issue_stage(0, 0);
    wait_async();
    __syncthreads();

    for (int s = 0; s < kNStage; ++s) {
      const int cur = s & 1;
      if (s + 1 < kNStage) issue_stage(s + 1, cur ^ 1);   // overlap next-chunk DMA

      const float* cA = sA + cur * kAbuf;
      const float* cB = sB + cur * kBbuf;
      #pragma unroll
      for (int c0 = 0; c0 < kKC; c0 += 4) {
        v2f a;
        a.x = cA[aBase + c0 * kSA];
        a.y = cA[aBase + c0 * kSA + kSA];
        const int bo = bBase + c0 * kSB;
        v2f b0, b1, b2, b3;
        b0.x = cB[bo +  0]; b0.y = cB[bo +  0 + kSB];
        b1.x = cB[bo + 16]; b1.y = cB[bo + 16 + kSB];
        b2.x = cB[bo + 32]; b2.y = cB[bo + 32 + kSB];
        b3.x = cB[bo + 48]; b3.y = cB[bo + 48 + kSB];
        acc0 = __builtin_amdgcn_wmma_f32_16x16x4_f32(false, a, false, b0, (short)0, acc0, false, false);
        acc1 = __builtin_amdgcn_wmma_f32_16x16x4_f32(false, a, false, b1, (short)0, acc1, false, false);
        acc2 = __builtin_amdgcn_wmma_f32_16x16x4_f32(false, a, false, b2, (short)0, acc2, false, false);
        acc3 = __builtin_amdgcn_wmma_f32_16x16x4_f32(false, a, false, b3, (short)0, acc3, false, false);
      }

      wait_async();        // next-stage DMA (this wave's) has landed
      __syncthreads();     // all waves done reading cur + all DMAs visible
    }

    // ---- extract the 21 even diagonals into sOut[dxIdx][x] ----
    // C/D layout: vgpr i, lane l -> M = i + 8*(l>>4), N = l&15
    const int xlb = hi << 3;
    #pragma unroll
    for (int j = 0; j < 4; ++j) {
      const v8f accj = (j == 0) ? acc0 : (j == 1) ? acc1 : (j == 2) ? acc2 : acc3;
      #pragma unroll
      for (int i = 0; i < 8; ++i) {
        int xl = xlb + i;                 // local x in tile
        int dx = (j << 4) - 24 + m - xl;  // x' - x
        if (dx >= -20 && dx <= 20 && ((dx & 1) == 0))
          sOut[((dx + 20) >> 1) * kW + x0 + xl] = accj[i] * (1.0f / 256.0f);
      }
    }
  }
  __syncthreads();

  // ---- contiguous store of 21 full rows (zeros if y2 out of range) ----
  for (int f = tid; f < kD * (kW / 4); f += 256) {
    int row = f >> 5;            // dxIdx
    int c4  = (f & 31) << 2;
    float4 v = rowValid ? *(const float4*)(sOut + row * kW + c4)
                        : make_float4(0.f, 0.f, 0.f, 0.f);
    size_t o = ((size_t)(b * (kD * kD) + dyIdx * kD + row) * kH + y) * kW + c4;
    *(float4*)(out + o) = v;
  }
}

extern "C" void kernel_launch(void* const* d_in, const int* in_sizes, int n_in,
                              void* d_out, int out_size, void* d_ws, size_t ws_size,
                              hipStream_t stream) {
  (void)in_sizes; (void)n_in; (void)d_ws; (void)ws_size; (void)out_size;
  const float* in1 = (const float*)d_in[0];
  const float* in2 = (const float*)d_in[1];
  float* out = (float*)d_out;
  dim3 grid(kH, kD, kB);   // (y, dyIdx, b) — every output element written each call
  corr_wmma_kernel<<<grid, 256, 0, stream>>>(in1, in2, out);
}